// MXGNet_38809324487233
// MI455X (gfx1250) — compile-verified
//
#include <hip/hip_runtime.h>
#include <math.h>
#include <stdint.h>

typedef __attribute__((ext_vector_type(2))) float v2f;
typedef __attribute__((ext_vector_type(4))) float v4f;
typedef __attribute__((ext_vector_type(8))) float v8f;
typedef __attribute__((ext_vector_type(4))) unsigned int v4u;
typedef __attribute__((ext_vector_type(8))) unsigned int v8u;

#define IN_DIM   128
#define T_STEPS  9
#define HIDN     64
#define MODF     32
#define SUMF     96
#define ATTN     12
#define OUT_DIM  64
#define NB       16               // batches per workgroup
#define ROWS     (NB * T_STEPS)   // 144 rows = 9 tiles of 16
#define NWAVES   9
#define BLOCK    (NWAVES * 32)    // 288 threads

// ---------------- LDS layout (bytes) ----------------
#define W1_STRIDE 140             // 140 % 64 == 12 -> conflict-free B-frag reads
#define OFF_W1    0
#define SZ_W1     (64 * W1_STRIDE * 4)          // 35840
#define M_STRIDE  68              // m_w^T [32][64+pad]
#define OFF_M     (OFF_W1 + SZ_W1)              // 35840
#define SZ_M      (32 * M_STRIDE * 4)           // 8704
#define X_STRIDE  140
#define OFF_X     (OFF_M + SZ_M)                // 44544 (16B aligned)
#define SZ_X      (NWAVES * 16 * X_STRIDE * 4)  // 80640
#define Y_STRIDE  33
#define OFF_Y     OFF_X           // overlay: Y written only after all GEMM1 X reads (barrier)
#define WF_STRIDE 204             // 204 % 64 == 12
#define OFF_WF    OFF_X           // overlay: Wf staged after branch loop
#define H_STRIDE  68
#define OFF_H     (OFF_X + SZ_X)                // 125184
#define SZ_H      (NWAVES * 16 * H_STRIDE * 4)  // 39168
#define FM_STRIDE 196
#define OFF_FM    (OFF_H + SZ_H)                // 164352
#define SZ_FM     (16 * FM_STRIDE * 4)          // 12544
#define S_STRIDE  100
#define OFF_S     (OFF_FM + SZ_FM)              // 176896
#define SZ_S      (16 * S_STRIDE * 4)           // 6400
#define OFF_CONST (OFF_S + SZ_S)                // 183296
#define SZ_CONST  (320 * 4)
#define SMEM_BYTES (OFF_CONST + SZ_CONST)       // 184576

// TDM pad codes for 128-dword rows padded to 140 dwords:
// pad_interval code 6 -> 128 DWORDs; pad_amount code 11 -> 12 DWORDs
#define PAD_I_128 6u
#define PAD_A_12  11u

__device__ __forceinline__ v8f wmma4(v2f a, v2f b, v8f c) {
    // D = A(16x4 f32) x B(4x16 f32) + C(16x16 f32)
    return __builtin_amdgcn_wmma_f32_16x16x4_f32(false, a, false, b,
                                                 (short)0, c, false, false);
}

// Issue one 2D Tensor-Data-Mover load: tensor (d0 x d1), tile (t0 x t1),
// row stride sd0 (all in elements), 4-byte elements, LDS pad per row.
__device__ __forceinline__ void tdm_load_2d(const void* gptr, void* lptr,
                                            unsigned d0, unsigned d1,
                                            unsigned t0, unsigned t1,
                                            unsigned sd0,
                                            unsigned pad_i, unsigned pad_a)
{
    unsigned long long ga = (unsigned long long)(size_t)gptr;
    unsigned lo = (unsigned)(unsigned long long)(size_t)lptr; // low 32b of generic = LDS byte offset
    v4u g0;
    g0[0] = 1u;                                            // count=1 (valid user D#)
    g0[1] = lo;                                            // lds_addr
    g0[2] = (unsigned)ga;                                  // global_addr[31:0]
    g0[3] = (unsigned)((ga >> 32) & 0x1FFFFFFull) | (2u << 30); // global_addr[56:32] | type=2
    v8u g1;
    g1[0] = (2u << 16) | (1u << 20) | (pad_i << 22) | (pad_a << 25); // 4B elems, pad_enable
    g1[1] = (d0 & 0xFFFFu) << 16;                          // tensor_dim0[15:0]
    g1[2] = ((d0 >> 16) & 0xFFFFu) | ((d1 & 0xFFFFu) << 16); // dim0[31:16] | dim1[15:0]
    g1[3] = ((d1 >> 16) & 0xFFFFu) | ((t0 & 0xFFFFu) << 16); // dim1[31:16] | tile_dim0
    g1[4] = (t1 & 0xFFFFu);                                // tile_dim1 (tile_dim2 = 0)
    g1[5] = sd0;                                           // tensor_dim0_stride[31:0]
    g1[6] = 0u;                                            // stride0 hi, stride1 lo (unused 2D)
    g1[7] = 0u;
    asm volatile("tensor_load_to_lds %0, %1" :: "s"(g0), "s"(g1) : "memory");
}

__global__ __launch_bounds__(BLOCK)
void mxg_fused(const float* __restrict__ fl02, const float* __restrict__ fl12,
               const float* __restrict__ pW1,  const float* __restrict__ pb1,
               const float* __restrict__ pg1,  const float* __restrict__ pbe1,
               const float* __restrict__ pmw,  const float* __restrict__ pmb,
               const float* __restrict__ pgm,  const float* __restrict__ pbem,
               const float* __restrict__ pWa,  const float* __restrict__ pba,
               const float* __restrict__ pWf,  const float* __restrict__ pbf,
               const float* __restrict__ pgf,  const float* __restrict__ pbef,
               float* __restrict__ out, int B)
{
    extern __shared__ char smem[];
    const int tid  = threadIdx.x;
    const int lane = tid & 31;
    const int w    = tid >> 5;       // wave id 0..8
    const int nlo  = lane & 15;
    const int half = lane >> 4;
    const int wg   = blockIdx.x;
    if (wg * NB >= B) return;

    float* W1L = (float*)(smem + OFF_W1);
    float* MLt = (float*)(smem + OFF_M);
    float* XL  = (float*)(smem + OFF_X);
    float* YL  = (float*)(smem + OFF_Y);
    float* WFL = (float*)(smem + OFF_WF);
    float* HL  = (float*)(smem + OFF_H);
    float* FML = (float*)(smem + OFF_FM);
    float* SL  = (float*)(smem + OFF_S);
    float* CL  = (float*)(smem + OFF_CONST);
    float* scale1 = CL;        float* shift1 = CL + 64;
    float* scale2 = CL + 128;  float* shift2 = CL + 160;
    float* scalef = CL + 192;  float* shiftf = CL + 256;

    // ---- wave 0: DMA W1 (64x128) and branch-0 inputs (144x128) into LDS via TDM ----
    if (w == 0) {
        tdm_load_2d(pW1, W1L, IN_DIM, HIDN, IN_DIM, HIDN, IN_DIM, PAD_I_128, PAD_A_12);
        tdm_load_2d(fl02 + (long)wg * ROWS * IN_DIM, XL,
                    IN_DIM, ROWS, IN_DIM, ROWS, IN_DIM, PAD_I_128, PAD_A_12);
        __builtin_amdgcn_s_wait_tensorcnt(0);
    }
    // ---- all waves: stage masked m_w^T and fused-BN constants ----
    for (int idx = tid; idx < HIDN * MODF; idx += BLOCK) {
        int j = idx >> 5, o = idx & 31;                       // m_w[j][o]
        float v = ((j >> 4) == (o >> 3)) ? pmw[idx] : 0.0f;   // block-diag mask
        MLt[o * M_STRIDE + j] = v;                            // store transposed
    }
    const float inv = rsqrtf(1.0f + 1e-5f);
    if (tid < 64)       { float s = pg1[tid] * inv; scale1[tid] = s; shift1[tid] = pb1[tid] * s + pbe1[tid]; }
    else if (tid < 96)  { int i = tid - 64; float s = pgm[i] * inv; scale2[i] = s; shift2[i] = pmb[i] * s + pbem[i]; }
    else if (tid < 160) { int i = tid - 96; float s = pgf[i] * inv; scalef[i] = s; shiftf[i] = pbf[i] * s + pbef[i]; }
    __syncthreads();

    for (int br = 0; br < 2; ++br) {
        if (br == 1) {
            // previous branch's Y readers finished at its last barrier; refill X via TDM
            if (w == 0) {
                tdm_load_2d(fl12 + (long)wg * ROWS * IN_DIM, XL,
                            IN_DIM, ROWS, IN_DIM, ROWS, IN_DIM, PAD_I_128, PAD_A_12);
                __builtin_amdgcn_s_wait_tensorcnt(0);
            }
            __syncthreads();
        }

        // ---- GEMM1: [16x128] @ W1^T[128x64] via f32 WMMA ----
        v8f a0 = {}, a1 = {}, a2 = {}, a3 = {};
        {
            const float* xr = XL + (w * 16 + nlo) * X_STRIDE;
            for (int ks = 0; ks < 32; ++ks) {
                int kk = ks * 4 + 2 * half;
                v2f a  = *(const v2f*)(xr + kk);
                v2f b0 = *(const v2f*)(W1L + ( 0 + nlo) * W1_STRIDE + kk);
                v2f b1 = *(const v2f*)(W1L + (16 + nlo) * W1_STRIDE + kk);
                v2f b2 = *(const v2f*)(W1L + (32 + nlo) * W1_STRIDE + kk);
                v2f b3 = *(const v2f*)(W1L + (48 + nlo) * W1_STRIDE + kk);
                a0 = wmma4(a, b0, a0);
                a1 = wmma4(a, b1, a1);
                a2 = wmma4(a, b2, a2);
                a3 = wmma4(a, b3, a3);
            }
        }
        // epilogue: bias+BN+ReLU fused -> H (per-wave region, same-wave consumer)
        {
            float* hw = HL + w * 16 * H_STRIDE;
            #pragma unroll
            for (int v = 0; v < 8; ++v) {
                int m = v + 8 * half;
                int c0 = nlo, c1 = 16 + nlo, c2 = 32 + nlo, c3 = 48 + nlo;
                hw[m * H_STRIDE + c0] = fmaxf(a0[v] * scale1[c0] + shift1[c0], 0.f);
                hw[m * H_STRIDE + c1] = fmaxf(a1[v] * scale1[c1] + shift1[c1], 0.f);
                hw[m * H_STRIDE + c2] = fmaxf(a2[v] * scale1[c2] + shift1[c2], 0.f);
                hw[m * H_STRIDE + c3] = fmaxf(a3[v] * scale1[c3] + shift1[c3], 0.f);
            }
        }

        // ---- GEMM2: [16x64] @ (m_w*mask)[64x32] via f32 WMMA ----
        v8f p0 = {}, p1 = {};
        {
            const float* hr = HL + (w * 16 + nlo) * H_STRIDE;
            for (int ks = 0; ks < 16; ++ks) {
                int kk = ks * 4 + 2 * half;
                v2f a  = *(const v2f*)(hr + kk);
                v2f b0 = *(const v2f*)(MLt + ( 0 + nlo) * M_STRIDE + kk);
                v2f b1 = *(const v2f*)(MLt + (16 + nlo) * M_STRIDE + kk);
                p0 = wmma4(a, b0, p0);
                p1 = wmma4(a, b1, p1);
            }
        }
        __syncthreads();   // all waves done reading X before Y overlays it
        {
            #pragma unroll
            for (int v = 0; v < 8; ++v) {
                int row = w * 16 + v + 8 * half;
                int c0 = nlo, c1 = 16 + nlo;
                YL[row * Y_STRIDE + c0] = fmaxf(p0[v] * scale2[c0] + shift2[c0], 0.f);
                YL[row * Y_STRIDE + c1] = fmaxf(p1[v] * scale2[c1] + shift2[c1], 0.f);
            }
        }
        __syncthreads();

        // ---- T-reduction: sum / mean / max over 9 steps ----
        for (int idx = tid; idx < NB * MODF; idx += BLOCK) {
            int bl = idx >> 5, f = idx & 31;
            const float* yb = YL + (bl * T_STEPS) * Y_STRIDE + f;
            float s = 0.f, mx = -3.4e38f;
            #pragma unroll
            for (int t = 0; t < T_STEPS; ++t) {
                float v = yb[t * Y_STRIDE];
                s += v; mx = fmaxf(mx, v);
            }
            SL[bl * S_STRIDE + f]      = s;
            SL[bl * S_STRIDE + 32 + f] = s * (1.0f / 9.0f);
            SL[bl * S_STRIDE + 64 + f] = mx;
        }
        __syncthreads();

        // ---- attention gate: 16 batches x 12 heads ----
        if (tid < NB * ATTN) {
            int bl = tid / ATTN, j = tid % ATTN;
            const float* sr = SL + bl * S_STRIDE;
            const float* wr = pWa + j * SUMF;
            float d = pba[j];
            for (int i = 0; i < SUMF; ++i) d += sr[i] * wr[i];
            float a = 1.0f / (1.0f + __expf(-d));
            float* fm = FML + bl * FM_STRIDE + br * SUMF + j * 8;
            #pragma unroll
            for (int i = 0; i < 8; ++i) fm[i] = sr[j * 8 + i] * a;
        }
        __syncthreads();
    }

    // ---- stage Wf (overlays X region; all X/Y readers finished).
    //      192-dword rows aren't a pow2 pad interval, so plain coalesced copy. ----
    for (int idx = tid; idx < (OUT_DIM * 2 * SUMF) / 4; idx += BLOCK) {
        int r = idx / 48, c4 = idx % 48;
        v4f v = *(const v4f*)(pWf + r * (2 * SUMF) + c4 * 4);
        *(v4f*)(WFL + r * WF_STRIDE + c4 * 4) = v;
    }
    __syncthreads();

    // ---- final GEMM: [16x192] @ Wf^T[192x64], waves 0..3 (one 16-col tile each) ----
    if (w < 4) {
        v8f acc = {};
        const float* fmr = FML + nlo * FM_STRIDE;
        const float* wfr = WFL + (w * 16 + nlo) * WF_STRIDE;
        for (int ks = 0; ks < 48; ++ks) {
            int kk = ks * 4 + 2 * half;
            v2f a = *(const v2f*)(fmr + kk);
            v2f b = *(const v2f*)(wfr + kk);
            acc = wmma4(a, b, acc);
        }
        #pragma unroll
        for (int v = 0; v < 8; ++v) {
            int m = v + 8 * half;
            int col = w * 16 + nlo;
            float o = fmaxf(acc[v] * scalef[col] + shiftf[col], 0.f);
            out[(long)(wg * NB + m) * OUT_DIM + col] = o;
        }
    }
}

extern "C" void kernel_launch(void* const* d_in, const int* in_sizes, int n_in,
                              void* d_out, int out_size, void* d_ws, size_t ws_size,
                              hipStream_t stream) {
    const float* fl02 = (const float*)d_in[0];
    const float* fl12 = (const float*)d_in[1];
    const float* W1   = (const float*)d_in[2];
    const float* b1   = (const float*)d_in[3];
    const float* g1   = (const float*)d_in[4];
    const float* be1  = (const float*)d_in[5];
    const float* m_w  = (const float*)d_in[6];
    const float* m_b  = (const float*)d_in[7];
    const float* gm   = (const float*)d_in[8];
    const float* bem  = (const float*)d_in[9];
    const float* Wa   = (const float*)d_in[10];
    const float* ba   = (const float*)d_in[11];
    const float* Wf   = (const float*)d_in[12];
    const float* bf   = (const float*)d_in[13];
    const float* gf   = (const float*)d_in[14];
    const float* bef  = (const float*)d_in[15];
    float* out = (float*)d_out;

    int B = in_sizes[0] / (T_STEPS * IN_DIM);
    int nwg = (B + NB - 1) / NB;

    (void)hipFuncSetAttribute((const void*)mxg_fused,
                              hipFuncAttributeMaxDynamicSharedMemorySize,
                              SMEM_BYTES);
    mxg_fused<<<nwg, BLOCK, SMEM_BYTES, stream>>>(
        fl02, fl12, W1, b1, g1, be1, m_w, m_b, gm, bem,
        Wa, ba, Wf, bf, gf, bef, out, B);
}